// XSA_50534585205467
// MI455X (gfx1250) — compile-verified
//
#include <hip/hip_runtime.h>
#include <hip/hip_bf16.h>

typedef __attribute__((ext_vector_type(2))) float v2f;
typedef __attribute__((ext_vector_type(8))) float v8f;

__device__ __forceinline__ v8f wmma_f32(v2f a, v2f b, v8f c) {
    // D = A(16x4,f32) * B(4x16,f32) + C(16x16,f32), full fp32 precision on the matrix pipe
    return __builtin_amdgcn_wmma_f32_16x16x4_f32(false, a, false, b, (short)0, c, false, false);
}

// ---------------------------------------------------------------------------
// GEMM: Y[M,N] = X[M,K] @ W[N,K]^T   (torch Linear convention)
// Wave computes a 64x64 tile (4x4 WMMA fragments -> 16 wmma per 8 loads).
// Block = 8 waves arranged 2(M) x 4(N) -> 128 x 256 block tile.
// A-frag and B-frag share the same per-lane pattern since both are [rows,K].
// ---------------------------------------------------------------------------
__global__ __launch_bounds__(256, 1) void gemm_xwT(const float* __restrict__ X,
                                                   const float* __restrict__ W,
                                                   float* __restrict__ Y,
                                                   int M, int N, int K) {
    const int lane = threadIdx.x & 31;
    const int wid  = threadIdx.x >> 5;
    const int l15  = lane & 15;
    const int hh   = lane >> 4;          // 0 or 1 (half-wave)
    const int m0 = blockIdx.y * 128 + (wid & 1) * 64;
    const int n0 = blockIdx.x * 256 + (wid >> 1) * 64;

    const float* xr[4];
    const float* wr[4];
#pragma unroll
    for (int i = 0; i < 4; ++i) {
        xr[i] = X + (size_t)(m0 + i * 16 + l15) * K + 2 * hh;
        wr[i] = W + (size_t)(n0 + i * 16 + l15) * K + 2 * hh;
    }

    v8f c[4][4];
#pragma unroll
    for (int i = 0; i < 4; ++i)
#pragma unroll
        for (int j = 0; j < 4; ++j) c[i][j] = (v8f){};

#pragma unroll 2
    for (int k0 = 0; k0 < K; k0 += 4) {
        v2f a[4], b[4];
#pragma unroll
        for (int i = 0; i < 4; ++i) a[i] = *(const v2f*)(xr[i] + k0);
#pragma unroll
        for (int j = 0; j < 4; ++j) b[j] = *(const v2f*)(wr[j] + k0);
#pragma unroll
        for (int i = 0; i < 4; ++i)
#pragma unroll
            for (int j = 0; j < 4; ++j) c[i][j] = wmma_f32(a[i], b[j], c[i][j]);
    }

#pragma unroll
    for (int i = 0; i < 4; ++i) {
#pragma unroll
        for (int r = 0; r < 8; ++r) {
            size_t row = (size_t)(m0 + i * 16 + r + 8 * hh);
            float* y = Y + row * N + n0 + l15;
#pragma unroll
            for (int j = 0; j < 4; ++j) y[j * 16] = c[i][j][r];
        }
    }
}

// ---------------------------------------------------------------------------
// RMSNorm of q and k rows in the qkv buffer [B,T,3,H,D], D=128.
// One wave per (b,t,s in {0,1},h) row: float4 per lane + wave reduction.
// ---------------------------------------------------------------------------
__global__ __launch_bounds__(256, 1) void rmsnorm_qk(float* __restrict__ qkv,
                                                     const float* __restrict__ qw,
                                                     const float* __restrict__ kw) {
    const int lane = threadIdx.x & 31;
    const int wid  = threadIdx.x >> 5;
    const int rid  = blockIdx.x * 8 + wid;   // 0 .. B*T*2*H-1
    const int h  = rid & 15;
    const int s  = (rid >> 4) & 1;
    const int bt = rid >> 5;

    float* row = qkv + (size_t)bt * 6144 + s * 2048 + h * 128;
    const float* wv = (s == 0) ? qw : kw;

    float4 v = *(const float4*)(row + lane * 4);
    float ss = v.x * v.x + v.y * v.y + v.z * v.z + v.w * v.w;
#pragma unroll
    for (int m = 16; m >= 1; m >>= 1) ss += __shfl_xor(ss, m, 32);
    const float sc = rsqrtf(ss * (1.0f / 128.0f) + 1.1920928955078125e-07f);

    float4 w4 = *(const float4*)(wv + lane * 4);
    v.x *= sc * w4.x; v.y *= sc * w4.y; v.z *= sc * w4.z; v.w *= sc * w4.w;
    *(float4*)(row + lane * 4) = v;
}

// ---------------------------------------------------------------------------
// Flash-style ALiBi attention. One wave owns a 16-row Q tile of one (b,h).
// S = (Q*scale) @ K^T via 32x f32 WMMA; online softmax per C-layout row;
// P converted C-layout -> A-layout through wave-private padded LDS;
// O += P @ V via 32x f32 WMMA. Epilogue: O/l + v_res_scale[h]*V.
// ---------------------------------------------------------------------------
__global__ __launch_bounds__(256, 1) void attn_kernel(const float* __restrict__ qkv,
                                                      const float* __restrict__ alibi,
                                                      const float* __restrict__ vres,
                                                      float* __restrict__ obuf) {
    const int T = 1024, D = 128, RS = 6144, C = 2048;
    __shared__ float pl[8][16 * 18];     // per-wave 16x16 P tile, stride 18 (8B aligned, bank-safe)

    const int lane = threadIdx.x & 31;
    const int wid  = threadIdx.x >> 5;
    const int w    = blockIdx.x * 8 + wid;   // 0 .. 8191
    const int i0   = (w & 63) << 4;          // q-tile start row
    const int bh   = w >> 6;
    const int h    = bh & 15;
    const int b    = bh >> 4;
    const int l15  = lane & 15;
    const int hh   = lane >> 4;

    const float* Qbase = qkv + (size_t)b * T * RS + h * D;        // s=0
    const float* Kbase = Qbase + C;                               // s=1
    const float* Vbase = Qbase + 2 * C;                           // s=2
    const float  scale = 0.08838834764831845f;                    // 128^-0.5

    // resident Q fragments, pre-scaled (32 k-steps of 4 over D=128)
    v2f qa[32];
    {
        const float* qrow = Qbase + (size_t)(i0 + l15) * RS + 2 * hh;
#pragma unroll
        for (int ks = 0; ks < 32; ++ks) qa[ks] = (*(const v2f*)(qrow + 4 * ks)) * scale;
    }

    v8f o[8];
#pragma unroll
    for (int nt = 0; nt < 8; ++nt) o[nt] = (v8f){};
    float mrow[8], lrow[8];
#pragma unroll
    for (int r = 0; r < 8; ++r) { mrow[r] = -3.0e38f; lrow[r] = 0.0f; }

    float* P = pl[wid];
    const float* abase = alibi + (size_t)h * T * T;

    for (int j0 = 0; j0 < T; j0 += 16) {
        // ---- S = Qs @ K^T (16x16 tile) ----
        v8f s = {};
        const float* krow = Kbase + (size_t)(j0 + l15) * RS + 2 * hh;
#pragma unroll
        for (int ks = 0; ks < 32; ++ks) {
            v2f kf = *(const v2f*)(krow + 4 * ks);
            s = wmma_f32(qa[ks], kf, s);
        }
        // ---- bias + online softmax; C-layout: VGPR r holds row (r+8*hh), col l15 ----
#pragma unroll
        for (int r = 0; r < 8; ++r) {
            const int qi = r + 8 * hh;
            float sv = s[r] + abase[(size_t)(i0 + qi) * T + j0 + l15];
            float mx = sv;
            mx = fmaxf(mx, __shfl_xor(mx, 1, 32));
            mx = fmaxf(mx, __shfl_xor(mx, 2, 32));
            mx = fmaxf(mx, __shfl_xor(mx, 4, 32));
            mx = fmaxf(mx, __shfl_xor(mx, 8, 32));
            const float mnew = fmaxf(mrow[r], mx);
            const float corr = __expf(mrow[r] - mnew);
            const float p    = __expf(sv - mnew);
            float rs = p;
            rs += __shfl_xor(rs, 1, 32);
            rs += __shfl_xor(rs, 2, 32);
            rs += __shfl_xor(rs, 4, 32);
            rs += __shfl_xor(rs, 8, 32);
            lrow[r] = lrow[r] * corr + rs;
            mrow[r] = mnew;
#pragma unroll
            for (int nt = 0; nt < 8; ++nt) o[nt][r] *= corr;
            P[qi * 18 + l15] = p;                 // stage P (C-layout) to LDS
        }
        asm volatile("s_wait_dscnt 0" ::: "memory");   // LDS turnaround (wave-private tile)
        // ---- reload P as A-layout fragments ----
        v2f pf[4];
#pragma unroll
        for (int ks = 0; ks < 4; ++ks)
            pf[ks] = *(const v2f*)(P + l15 * 18 + 4 * ks + 2 * hh);
        asm volatile("" ::: "memory");
        // ---- O += P @ V  (8 N-tiles over D=128) ----
#pragma unroll
        for (int ks = 0; ks < 4; ++ks) {
            const int kb = 4 * ks + 2 * hh;
            const float* vr0 = Vbase + (size_t)(j0 + kb) * RS + l15;
            const float* vr1 = vr0 + RS;
#pragma unroll
            for (int nt = 0; nt < 8; ++nt) {
                v2f vf;
                vf[0] = vr0[nt * 16];
                vf[1] = vr1[nt * 16];
                o[nt] = wmma_f32(pf[ks], vf, o[nt]);
            }
        }
    }

    // ---- epilogue: normalize, v-residual, store [B,T,H*D] ----
    const float vrs = vres[h];
#pragma unroll
    for (int r = 0; r < 8; ++r) {
        const int qi = i0 + r + 8 * hh;
        const float linv = 1.0f / lrow[r];
        const float* vrow = Vbase + (size_t)qi * RS;
        float* orow = obuf + (size_t)(b * T + qi) * C + h * D;
#pragma unroll
        for (int nt = 0; nt < 8; ++nt)
            orow[nt * 16 + l15] = o[nt][r] * linv + vrs * vrow[nt * 16 + l15];
    }
}

// ---------------------------------------------------------------------------
extern "C" void kernel_launch(void* const* d_in, const int* in_sizes, int n_in,
                              void* d_out, int out_size, void* d_ws, size_t ws_size,
                              hipStream_t stream) {
    (void)in_sizes; (void)n_in; (void)out_size;
    const float* x     = (const float*)d_in[0];  // [8,1024,2048]
    const float* alibi = (const float*)d_in[1];  // [1,16,1024,1024]
    const float* Wqkv  = (const float*)d_in[2];  // [6144,2048]
    const float* Wout  = (const float*)d_in[3];  // [2048,2048]
    const float* vres  = (const float*)d_in[4];  // [16]
    const float* qw    = (const float*)d_in[5];  // [128]
    const float* kw    = (const float*)d_in[6];  // [128]
    float* out = (float*)d_out;                  // [8,1024,2048]

    const size_t qkv_elems  = (size_t)8 * 1024 * 6144;   // 201 MB
    const size_t obuf_elems = (size_t)8 * 1024 * 2048;   // 67 MB
    if (ws_size < (qkv_elems + obuf_elems) * sizeof(float)) return;
    float* qkv  = (float*)d_ws;
    float* obuf = qkv + qkv_elems;

    // 1) QKV projection: [8192,2048] @ [6144,2048]^T -> [8192,6144]
    gemm_xwT<<<dim3(6144 / 256, 8192 / 128), 256, 0, stream>>>(x, Wqkv, qkv, 8192, 6144, 2048);
    // 2) RMSNorm q and k per head row (B*T*2*H rows of 128)
    rmsnorm_qk<<<(8 * 1024 * 2 * 16) / 8, 256, 0, stream>>>(qkv, qw, kw);
    // 3) ALiBi attention + v-residual -> [B,T,C]
    attn_kernel<<<(8 * 16 * 64) / 8, 256, 0, stream>>>(qkv, alibi, vres, obuf);
    // 4) Output projection: [8192,2048] @ [2048,2048]^T -> d_out
    gemm_xwT<<<dim3(2048 / 256, 8192 / 128), 256, 0, stream>>>(obuf, Wout, out, 8192, 2048, 2048);
}